// SkelEmbeddingScore_3229815406607
// MI455X (gfx1250) — compile-verified
//
#include <hip/hip_runtime.h>
#include <hip/hip_bf16.h>

#define NT   768      // frames (n == m == 768)
#define LAT  256      // latent dim
#define CIN  150      // NJ*NF = 25*6
#define BIGV 1.0e8f
#define NDIAG 1535    // NT + NT - 1 anti-diagonals of D

typedef __attribute__((ext_vector_type(2))) float v2f;
typedef __attribute__((ext_vector_type(8))) float v8f;

// ---------------------------------------------------------------------------
// K1: embed  ex[t][l] = sum_c x[c*NT + t] * W[l*CIN + c] + b[l]
// grid = 2*NT blocks of LAT threads; block -> (tensor, t), thread -> l
// ---------------------------------------------------------------------------
__global__ void embed_kernel(const float* __restrict__ x,
                             const float* __restrict__ y,
                             const float* __restrict__ W,
                             const float* __restrict__ b,
                             float* __restrict__ ex,
                             float* __restrict__ ey) {
    const int which = (blockIdx.x >= NT) ? 1 : 0;
    const int t = blockIdx.x - which * NT;
    const int l = threadIdx.x;
    const float* __restrict__ src = which ? y : x;
    float* __restrict__ dst = which ? ey : ex;

    float acc = b[l];
#pragma unroll 5
    for (int c = 0; c < CIN; ++c) {
        acc = fmaf(src[c * NT + t], W[l * CIN + c], acc);
    }
    dst[t * LAT + l] = acc;
}

// ---------------------------------------------------------------------------
// K2: row squared-norms of ex / ey
// ---------------------------------------------------------------------------
__global__ void norm_kernel(const float* __restrict__ ex,
                            const float* __restrict__ ey,
                            float* __restrict__ nx,
                            float* __restrict__ ny) {
    const int gid = blockIdx.x * blockDim.x + threadIdx.x;  // 0 .. 2*NT-1
    const int which = (gid >= NT) ? 1 : 0;
    const int row = gid - which * NT;
    const float4* __restrict__ p =
        (const float4*)((which ? ey : ex) + (size_t)row * LAT);
    float s = 0.0f;
#pragma unroll 4
    for (int i = 0; i < LAT / 4; ++i) {
        float4 v = p[i];
        s = fmaf(v.x, v.x, s);
        s = fmaf(v.y, v.y, s);
        s = fmaf(v.z, v.z, s);
        s = fmaf(v.w, v.w, s);
    }
    (which ? ny : nx)[row] = s;
}

// ---------------------------------------------------------------------------
// K3: D tile via V_WMMA_F32_16X16X4_F32, written into anti-diagonal-skewed
// layout Dsk[(i+j)*NT + i] = nx[i] + ny[j] - 2 * dot(ex[i], ey[j]).
// One wave32 per 16x16 tile.  ISA A-layout (32-bit A 16x4): lane L holds
// row M=(L&15), K = 2*(L>>4) in v0 and K+1 in v1.  B (4x16) mirrors: lane L
// holds col N=(L&15), rows K=2*(L>>4), K+1.  Since B[k][n] = ey[j0+n][k],
// both operands load a float2 at (row*LAT + k0 + 2*(L>>4)).
// ---------------------------------------------------------------------------
__global__ void pairD_kernel(const float* __restrict__ ex,
                             const float* __restrict__ ey,
                             const float* __restrict__ nx,
                             const float* __restrict__ ny,
                             float* __restrict__ Dsk) {
    const int j0 = blockIdx.x * 16;
    const int i0 = blockIdx.y * 16;
    const int lane  = threadIdx.x;          // 0..31, EXEC all ones (no divergence)
    const int mrow  = lane & 15;
    const int khalf = (lane >> 4) << 1;     // 0 or 2

    const float* __restrict__ pa = ex + (size_t)(i0 + mrow) * LAT + khalf;
    const float* __restrict__ pb = ey + (size_t)(j0 + mrow) * LAT + khalf;

    v8f c = {0.f, 0.f, 0.f, 0.f, 0.f, 0.f, 0.f, 0.f};
#pragma unroll 4
    for (int k0 = 0; k0 < LAT; k0 += 4) {
        v2f a  = *(const v2f*)(pa + k0);
        v2f bb = *(const v2f*)(pb + k0);
        // 8 args: (neg_a, A, neg_b, B, c_mod, C, reuse_a, reuse_b)
        c = __builtin_amdgcn_wmma_f32_16x16x4_f32(false, a, false, bb,
                                                  (short)0, c, false, false);
    }

    // C/D layout: VGPR r holds M = r + 8*(lane>=16), N = lane & 15.
    const int j = j0 + mrow;
    const float nyv = ny[j];
    const int mbase = i0 + ((lane >> 4) << 3);
#pragma unroll
    for (int r = 0; r < 8; ++r) {
        const int i = mbase + r;
        const float d = nx[i] + nyv - 2.0f * c[r];
        Dsk[(size_t)(i + j) * NT + i] = d;
    }
}

// ---------------------------------------------------------------------------
// K4: soft-DTW wavefront.  Single workgroup, 3 rotating LDS diagonals,
// one barrier per anti-diagonal (1535 sequential steps), gamma == 1.
// R[i,j] = D[i-1,j-1] + softmin(R[i-1,j-1], R[i-1,j], R[i,j-1])
// softmin(a,b,c) = min - log(exp(min-a)+exp(min-b)+exp(min-c))
// ---------------------------------------------------------------------------
__global__ void softdtw_kernel(const float* __restrict__ Dsk,
                               float* __restrict__ out) {
    __shared__ float diag[3][NT + 1];

    const int t = threadIdx.x;              // 0..799; slots 0..768 active
    if (t <= NT) {
        diag[0][t] = (t == 0) ? 0.0f : BIGV;   // k = 0 diagonal
        diag[1][t] = BIGV;                     // k = 1 diagonal
    }

    for (int k = 2; k <= 2 * NT; ++k) {
        __syncthreads();
        const int cur  = k % 3;
        const int dm1i = (k - 1) % 3;
        const int dm2i = (k - 2) % 3;
        if (t <= NT) {
            const int j = k - t;
            const bool valid = (t >= 1) && (j >= 1) && (j <= NT);
            float nv = BIGV;
            if (valid) {
                const float a = diag[dm2i][t - 1];  // R[i-1, j-1]
                const float b = diag[dm1i][t - 1];  // R[i-1, j]
                const float cl = diag[dm1i][t];     // R[i,   j-1]
                const float mn = fminf(a, fminf(b, cl));
                const float s = expf(mn - a) + expf(mn - b) + expf(mn - cl);
                const float smin = mn - logf(s);
                // D[i-1, j-1] at skewed row (i-1)+(j-1) = k-2, column i-1 = t-1
                nv = Dsk[(size_t)(k - 2) * NT + (t - 1)] + smin;
            }
            diag[cur][t] = nv;
        }
    }
    __syncthreads();
    if (t == 0) out[0] = diag[(2 * NT) % 3][NT];   // R[n, m]
}

// ---------------------------------------------------------------------------
// Launch
// inputs: x[1,25,6,768] f32, y[...] f32, W[256,150] f32, b[256] f32
// output: 1 float
// workspace: ex | ey | nx | ny | Dsk  (~6.3 MB)
// ---------------------------------------------------------------------------
extern "C" void kernel_launch(void* const* d_in, const int* in_sizes, int n_in,
                              void* d_out, int out_size, void* d_ws, size_t ws_size,
                              hipStream_t stream) {
    const float* x = (const float*)d_in[0];
    const float* y = (const float*)d_in[1];
    const float* W = (const float*)d_in[2];
    const float* b = (const float*)d_in[3];
    float* out = (float*)d_out;

    char* ws = (char*)d_ws;
    float* ex  = (float*)(ws);                                    // 768*256 f32
    float* ey  = (float*)(ws + (size_t)NT * LAT * 4);             // 768*256 f32
    float* nx  = (float*)(ws + (size_t)2 * NT * LAT * 4);         // 768 f32
    float* ny  = (float*)(ws + (size_t)2 * NT * LAT * 4 + 3072);  // 768 f32
    float* Dsk = (float*)(ws + (size_t)2 * NT * LAT * 4 + 6144);  // 1535*768 f32

    embed_kernel<<<dim3(2 * NT), dim3(LAT), 0, stream>>>(x, y, W, b, ex, ey);
    norm_kernel<<<dim3(2 * NT / 256), dim3(256), 0, stream>>>(ex, ey, nx, ny);
    pairD_kernel<<<dim3(NT / 16, NT / 16), dim3(32), 0, stream>>>(ex, ey, nx, ny, Dsk);
    softdtw_kernel<<<dim3(1), dim3(800), 0, stream>>>(Dsk, out);
}